// GATDecoder_28432683500205
// MI455X (gfx1250) — compile-verified
//
#include <hip/hip_runtime.h>
#include <hip/hip_bf16.h>

#define N_NODES 50000
#define N_EDGES 800000
#define HEADS   4

typedef __attribute__((ext_vector_type(2))) float v2f;
typedef __attribute__((ext_vector_type(8))) float v8f;

// ---------------------------------------------------------------------------
// GEMM: Y[M,N] = X[M,K] @ W[K,N] with V_WMMA_F32_16X16X4_F32.
// One wave -> 16(M) x 64(N) strip: A fragment reused across 4 accumulators.
// M%16==0, N%64==0, K%4==0 for all three layers (3125x16, {256,128}, {64,256,128}).
// A layout (16x4 f32): lanes 0-15 hold {K=k,k+1} for M=lane; lanes 16-31 {k+2,k+3}.
// B layout (4x16 f32): VGPR0 lanes0-15=K row k, lanes16-31=row k+2; VGPR1 rows k+1/k+3.
// C/D: 8 VGPRs, lane group selects M half, N = lane&15.
// ---------------------------------------------------------------------------
__global__ __launch_bounds__(256) void gemm_wmma_f32(
    const float* __restrict__ X, const float* __restrict__ W,
    float* __restrict__ Y, int M, int K, int N) {
  const int mt = M >> 4;
  const int nt = N >> 6;
  int wave = (blockIdx.x * blockDim.x + threadIdx.x) >> 5;
  if (wave >= mt * nt) return;              // wave-uniform: EXEC stays all-ones
  int tm = wave % mt;
  int tn = wave / mt;
  int m0 = tm << 4;
  int n0 = tn << 6;
  int lane = threadIdx.x & 31;
  int hi   = lane >> 4;                     // 0: K pair {0,1}; 1: {2,3}
  int lm   = lane & 15;

  v8f c0 = {}, c1 = {}, c2 = {}, c3 = {};
  const float* xrow = X + (size_t)(m0 + lm) * K;
  for (int k = 0; k < K; k += 4) {
    int kk = k + (hi << 1);
    v2f a = *(const v2f*)(xrow + kk);       // aligned 8B (K multiple of 4)
    const float* w0 = W + (size_t)kk * N + n0 + lm;
    const float* w1 = w0 + N;
    v2f b0 = { w0[0],  w1[0]  };
    v2f b1 = { w0[16], w1[16] };
    v2f b2 = { w0[32], w1[32] };
    v2f b3 = { w0[48], w1[48] };
    c0 = __builtin_amdgcn_wmma_f32_16x16x4_f32(false, a, false, b0, (short)0, c0, false, false);
    c1 = __builtin_amdgcn_wmma_f32_16x16x4_f32(false, a, false, b1, (short)0, c1, false, false);
    c2 = __builtin_amdgcn_wmma_f32_16x16x4_f32(false, a, false, b2, (short)0, c2, false, false);
    c3 = __builtin_amdgcn_wmma_f32_16x16x4_f32(false, a, false, b3, (short)0, c3, false, false);
  }
  float* yrow = Y + (size_t)(m0 + (hi << 3)) * N + n0 + lm;
  for (int i = 0; i < 8; ++i) {
    float* yp = yrow + (size_t)i * N;
    yp[0]  = c0[i];
    yp[16] = c1[i];
    yp[32] = c2[i];
    yp[48] = c3[i];
  }
}

// ---------------------------------------------------------------------------
// per-(node,head) attention logits: al = sum_c h[n,h,c] * a[h,c]
// ---------------------------------------------------------------------------
__global__ void alpha_kernel(const float* __restrict__ h,
                             const float* __restrict__ a_src,
                             const float* __restrict__ a_dst,
                             float* __restrict__ al_s, float* __restrict__ al_d,
                             int Nn, int C) {
  int idx = blockIdx.x * blockDim.x + threadIdx.x;
  if (idx >= Nn * HEADS) return;
  int hh = idx % HEADS;
  const float* hp = h + (size_t)idx * C;
  const float* as = a_src + hh * C;
  const float* ad = a_dst + hh * C;
  float ss = 0.f, sd = 0.f;
  for (int c = 0; c < C; ++c) { float v = hp[c]; ss += v * as[c]; sd += v * ad[c]; }
  al_s[idx] = ss;
  al_d[idx] = sd;
}

__global__ void fill_u32(unsigned* __restrict__ p, unsigned v, int n) {
  int i = blockIdx.x * blockDim.x + threadIdx.x;
  if (i < n) p[i] = v;
}

// monotone float->uint key: key order == float order; key 0 acts as -inf sentinel
__device__ __forceinline__ unsigned f2key(float f) {
  unsigned b = __float_as_uint(f);
  return (b & 0x80000000u) ? ~b : (b | 0x80000000u);
}
__device__ __forceinline__ float key2f(unsigned k) {
  return __uint_as_float((k & 0x80000000u) ? (k & 0x7fffffffu) : ~k);
}

// pass A: e = leaky_relu(al_s[src] + al_d[dst]); segment max via atomicMax on key
__global__ void edge_logits(const long long* __restrict__ esrc,
                            const long long* __restrict__ edst,
                            const float* __restrict__ al_s,
                            const float* __restrict__ al_d,
                            float* __restrict__ ebuf, unsigned* __restrict__ mkey) {
  int idx = blockIdx.x * blockDim.x + threadIdx.x;
  if (idx >= N_EDGES * HEADS) return;
  int e = idx / HEADS, hh = idx % HEADS;
  int s = (int)esrc[e], d = (int)edst[e];
  float v = al_s[s * HEADS + hh] + al_d[d * HEADS + hh];
  v = v > 0.f ? v : 0.2f * v;
  ebuf[idx] = v;
  atomicMax(&mkey[d * HEADS + hh], f2key(v));
}

// pass B: ex = exp(e - m[dst]); segment sum
__global__ void edge_exp(const long long* __restrict__ edst,
                         const unsigned* __restrict__ mkey,
                         float* __restrict__ ebuf, float* __restrict__ ssum) {
  int idx = blockIdx.x * blockDim.x + threadIdx.x;
  if (idx >= N_EDGES * HEADS) return;
  int e = idx / HEADS, hh = idx % HEADS;
  int d = (int)edst[e];
  float m = key2f(mkey[d * HEADS + hh]);
  float ex = __expf(ebuf[idx] - m);
  ebuf[idx] = ex;
  atomicAdd(&ssum[d * HEADS + hh], ex);
}

// pass C: out[dst] += alpha * h[src]; one thread per (edge, head, 4-wide c group)
__global__ void edge_scatter(const long long* __restrict__ esrc,
                             const long long* __restrict__ edst,
                             const float* __restrict__ hfeat,
                             const float* __restrict__ ebuf,
                             const float* __restrict__ ssum,
                             float* __restrict__ out, int C) {
  int cg = C >> 2;
  int per = HEADS * cg;
  long long idx = (long long)blockIdx.x * blockDim.x + threadIdx.x;
  if (idx >= (long long)N_EDGES * per) return;
  int t = (int)(idx % per);
  int e = (int)(idx / per);
  int hh = t / cg;
  int c4 = (t % cg) << 2;
  int s = (int)esrc[e], d = (int)edst[e];
  float alpha = ebuf[e * HEADS + hh] / (ssum[d * HEADS + hh] + 1e-16f);
  const float4 v = *(const float4*)(hfeat + ((size_t)s * HEADS + hh) * C + c4);
  float* op = out + ((size_t)d * HEADS + hh) * C + c4;
  atomicAdd(op + 0, alpha * v.x);
  atomicAdd(op + 1, alpha * v.y);
  atomicAdd(op + 2, alpha * v.z);
  atomicAdd(op + 3, alpha * v.w);
}

__global__ void add_bias(float* __restrict__ out, const float* __restrict__ b,
                         int total, int F) {
  int i = blockIdx.x * blockDim.x + threadIdx.x;
  if (i < total) out[i] += b[i % F];
}

// layer 3: mean over heads + bias -> final output [N, C]
__global__ void head_mean_bias(const float* __restrict__ acc,
                               const float* __restrict__ b,
                               float* __restrict__ out, int C) {
  int i = blockIdx.x * blockDim.x + threadIdx.x;
  if (i >= N_NODES * C) return;
  int n = i / C, c = i % C;
  float s = 0.f;
  for (int h = 0; h < HEADS; ++h) s += acc[((size_t)n * HEADS + h) * C + c];
  out[i] = s * (1.f / HEADS) + b[c];
}

static inline int cdiv(long long a, int b) { return (int)((a + b - 1) / b); }

extern "C" void kernel_launch(void* const* d_in, const int* in_sizes, int n_in,
                              void* d_out, int out_size, void* d_ws, size_t ws_size,
                              hipStream_t stream) {
  const float*      x    = (const float*)d_in[0];
  const long long*  ei   = (const long long*)d_in[1];   // int64 in reference
  const float* W1   = (const float*)d_in[2];
  const float* a1s  = (const float*)d_in[3];
  const float* a1d  = (const float*)d_in[4];
  const float* b1   = (const float*)d_in[5];
  const float* W2   = (const float*)d_in[6];
  const float* a2s  = (const float*)d_in[7];
  const float* a2d  = (const float*)d_in[8];
  const float* b2   = (const float*)d_in[9];
  const float* W3   = (const float*)d_in[10];
  const float* a3s  = (const float*)d_in[11];
  const float* a3d  = (const float*)d_in[12];
  const float* b3   = (const float*)d_in[13];
  float* out = (float*)d_out;

  const long long* esrc = ei;
  const long long* edst = ei + N_EDGES;

  // workspace carve-up (~119 MB)
  char* ws = (char*)d_ws;
  size_t off = 0;
  float*    bufA = (float*)(ws + off); off += (size_t)N_NODES * 256 * 4;   // h per layer
  float*    bufB = (float*)(ws + off); off += (size_t)N_NODES * 256 * 4;   // input / accum
  float*    al_s = (float*)(ws + off); off += (size_t)N_NODES * HEADS * 4;
  float*    al_d = (float*)(ws + off); off += (size_t)N_NODES * HEADS * 4;
  unsigned* mkey = (unsigned*)(ws + off); off += (size_t)N_NODES * HEADS * 4;
  float*    ssum = (float*)(ws + off); off += (size_t)N_NODES * HEADS * 4;
  float*    ebuf = (float*)(ws + off); off += (size_t)N_EDGES * HEADS * 4;

  const int nh = N_NODES * HEADS;
  const int eh = N_EDGES * HEADS;

  auto run_layer = [&](const float* xin, int Fin, const float* Wm,
                       const float* as, const float* ad, int C,
                       float* hbuf, float* outbuf) {
    int Fout = HEADS * C;
    int waves = (N_NODES / 16) * (Fout / 64);
    gemm_wmma_f32<<<cdiv((long long)waves * 32, 256), 256, 0, stream>>>(
        xin, Wm, hbuf, N_NODES, Fin, Fout);
    alpha_kernel<<<cdiv(nh, 256), 256, 0, stream>>>(hbuf, as, ad, al_s, al_d,
                                                    N_NODES, C);
    // stream-ordered: gemm has consumed xin (may alias outbuf) before zeroing
    fill_u32<<<cdiv(nh, 256), 256, 0, stream>>>(mkey, 0u, nh);
    fill_u32<<<cdiv(nh, 256), 256, 0, stream>>>((unsigned*)ssum, 0u, nh);
    fill_u32<<<cdiv((long long)N_NODES * Fout, 256), 256, 0, stream>>>(
        (unsigned*)outbuf, 0u, N_NODES * Fout);
    edge_logits<<<cdiv(eh, 256), 256, 0, stream>>>(esrc, edst, al_s, al_d, ebuf, mkey);
    edge_exp<<<cdiv(eh, 256), 256, 0, stream>>>(edst, mkey, ebuf, ssum);
    edge_scatter<<<cdiv((long long)N_EDGES * HEADS * (C >> 2), 256), 256, 0, stream>>>(
        esrc, edst, hbuf, ebuf, ssum, outbuf, C);
  };

  // layer 1: 64 -> 4x64 concat (256)
  run_layer(x, 64, W1, a1s, a1d, 64, bufA, bufB);
  add_bias<<<cdiv((long long)N_NODES * 256, 256), 256, 0, stream>>>(bufB, b1,
                                                                    N_NODES * 256, 256);
  // layer 2: 256 -> 4x32 concat (128)
  run_layer(bufB, 256, W2, a2s, a2d, 32, bufA, bufB);
  add_bias<<<cdiv((long long)N_NODES * 128, 256), 256, 0, stream>>>(bufB, b2,
                                                                    N_NODES * 128, 128);
  // layer 3 (recon): 128 -> 4x64, mean over heads -> 64
  run_layer(bufB, 128, W3, a3s, a3d, 64, bufA, bufB);
  head_mean_bias<<<cdiv((long long)N_NODES * 64, 256), 256, 0, stream>>>(bufB, b3,
                                                                         out, 64);
}